// Mutual_Attention_1451698946249
// MI455X (gfx1250) — compile-verified
//
#include <hip/hip_runtime.h>
#include <hip/hip_bf16.h>
#include <math.h>

typedef __attribute__((ext_vector_type(2))) float v2f;
typedef __attribute__((ext_vector_type(8))) float v8f;

#define B_  8
#define C_  192
#define HW_ 16384
#define NH_ 6
#define CH_ 32

__device__ __forceinline__ v8f wmma_f32(v2f a, v2f b, v8f c) {
  // V_WMMA_F32_16X16X4_F32 : D = A(16x4) * B(4x16) + C(16x16), fp32
  return __builtin_amdgcn_wmma_f32_16x16x4_f32(
      /*neg_a=*/false, a, /*neg_b=*/false, b,
      /*c_mod=*/(short)0, c, /*reuse_a=*/false, /*reuse_b=*/false);
}

// dst[z][o][n] = sum_c W[o][c] * src[z][c][n]   (1x1 conv == GEMM, M=192,K=192,N=16384)
// block: 256 thr = 8 waves; tile 32(M) x 256(N); wave = 16(M) x 64(N)
__global__ __launch_bounds__(256)
void proj_gemm(const float* __restrict__ W, const float* __restrict__ src,
               float* __restrict__ dst) {
  __shared__ float Ws[32 * 196];              // stride 196 (mod 64 == 4) -> conflict-free cols
  const int tid  = threadIdx.x;
  const int lane = tid & 31;
  const int w    = tid >> 5;
  const int z      = blockIdx.z;
  const int mBlock = blockIdx.y * 32;
  const int nBlock = blockIdx.x * 256;

  for (int i = tid; i < 32 * C_; i += 256) {
    int r = i / C_, c = i - r * C_;
    Ws[r * 196 + c] = W[(mBlock + r) * C_ + c];
  }
  __syncthreads();

  const int wm = w >> 2, wn = w & 3;
  const int mBase = wm * 16;                  // LDS row offset
  const int nBase = nBlock + wn * 64;
  const int hl  = lane >> 4;                  // lane half selects K=0,1 vs K=2,3
  const int l15 = lane & 15;

  const float* srcB = src + (size_t)z * C_ * HW_;
  v8f acc0 = {}, acc1 = {}, acc2 = {}, acc3 = {};
  for (int k0 = 0; k0 < C_; k0 += 4) {
    const int ks = k0 + 2 * hl;
    v2f a;
    a.x = Ws[(mBase + l15) * 196 + ks];
    a.y = Ws[(mBase + l15) * 196 + ks + 1];
    const float* p = srcB + (size_t)ks * HW_ + nBase + l15;
    v2f b0, b1, b2, b3;
    b0.x = p[0];  b0.y = p[HW_];
    b1.x = p[16]; b1.y = p[HW_ + 16];
    b2.x = p[32]; b2.y = p[HW_ + 32];
    b3.x = p[48]; b3.y = p[HW_ + 48];
    acc0 = wmma_f32(a, b0, acc0);
    acc1 = wmma_f32(a, b1, acc1);
    acc2 = wmma_f32(a, b2, acc2);
    acc3 = wmma_f32(a, b3, acc3);
  }
  float* dstB = dst + (size_t)z * C_ * HW_;
  const int rowBase = mBlock + mBase + hl * 8;
#pragma unroll
  for (int vv = 0; vv < 8; ++vv) {
    float* dp = dstB + (size_t)(rowBase + vv) * HW_ + nBase + l15;
    dp[0]  = acc0[vv];
    dp[16] = acc1[vv];
    dp[32] = acc2[vv];
    dp[48] = acc3[vv];
  }
}

// inv_norm[r] = 1 / max(||row_r||_2, 1e-12), rows: B*C of q then B*C of k
__global__ __launch_bounds__(256)
void row_norms(const float* __restrict__ q, const float* __restrict__ k,
               float* __restrict__ inv_norm) {
  __shared__ float red[256];
  const int r  = blockIdx.x;
  const int NR = B_ * C_;
  const float* base = (r < NR) ? (q + (size_t)r * HW_) : (k + (size_t)(r - NR) * HW_);
  float s = 0.f;
  for (int i = threadIdx.x; i < HW_; i += 256) {
    float t = base[i];
    s += t * t;
  }
  red[threadIdx.x] = s;
  __syncthreads();
  for (int off = 128; off > 0; off >>= 1) {
    if (threadIdx.x < off) red[threadIdx.x] += red[threadIdx.x + off];
    __syncthreads();
  }
  if (threadIdx.x == 0) {
    float n = fmaxf(sqrtf(red[0]), 1e-12f);
    inv_norm[r] = 1.0f / n;
  }
}

// attn[bh] = softmax( (q_n . k_n^T) * invq * invk * temp ), 32x32 per (b,head)
// 8 waves: 4 quadrants x 2 K-slices, K=16384 staged 64 at a time in LDS
__global__ __launch_bounds__(256)
void attn_kernel(const float* __restrict__ q, const float* __restrict__ k,
                 const float* __restrict__ inv_norm, const float* __restrict__ temp,
                 float* __restrict__ attn) {
  constexpr int TK = 64, LD = TK + 1;         // stride 65 -> conflict-free columns
  __shared__ float qs[32 * LD];
  __shared__ float ks_[32 * LD];
  __shared__ float Sred[8 * 16 * 16];
  __shared__ float S[32 * 33];

  const int tid  = threadIdx.x;
  const int lane = tid & 31;
  const int w    = tid >> 5;
  const int bh = blockIdx.x;
  const int b  = bh / NH_;
  const int h  = bh - b * NH_;

  const float* qB = q + ((size_t)b * C_ + h * CH_) * HW_;
  const float* kB = k + ((size_t)b * C_ + h * CH_) * HW_;

  const int quad = w & 3;
  const int cHalf = quad >> 1, dHalf = quad & 1;
  const int kbase = (w >> 2) * 32;            // K-slice within staged tile
  const int hl = lane >> 4, l15 = lane & 15;

  v8f acc = {};
  for (int n0 = 0; n0 < HW_; n0 += TK) {
    for (int i = tid; i < 32 * TK; i += 256) {
      int r = i >> 6, c = i & 63;
      qs[r * LD + c]  = qB[(size_t)r * HW_ + n0 + c];
      ks_[r * LD + c] = kB[(size_t)r * HW_ + n0 + c];
    }
    __syncthreads();
#pragma unroll
    for (int kk = 0; kk < 32; kk += 4) {
      const int ksel = kbase + kk + 2 * hl;
      v2f a, bb;
      a.x  = qs[(cHalf * 16 + l15) * LD + ksel];
      a.y  = qs[(cHalf * 16 + l15) * LD + ksel + 1];
      bb.x = ks_[(dHalf * 16 + l15) * LD + ksel];
      bb.y = ks_[(dHalf * 16 + l15) * LD + ksel + 1];
      acc = wmma_f32(a, bb, acc);
    }
    __syncthreads();
  }
#pragma unroll
  for (int vv = 0; vv < 8; ++vv)
    Sred[(w * 16 + vv + hl * 8) * 16 + l15] = acc[vv];
  __syncthreads();

  const float t = temp[h];
  const float* invq = inv_norm + b * C_ + h * CH_;
  const float* invk = inv_norm + B_ * C_ + b * C_ + h * CH_;
  for (int i = tid; i < 1024; i += 256) {
    int c = i >> 5, d = i & 31;
    int qd = (c >> 4) * 2 + (d >> 4);
    float val = Sred[(qd * 16 + (c & 15)) * 16 + (d & 15)]
              + Sred[((qd + 4) * 16 + (c & 15)) * 16 + (d & 15)];
    S[c * 33 + d] = val * invq[c] * invk[d] * t;
  }
  __syncthreads();

  if (tid < 32) {                              // softmax over rows of 32 (tiny)
    float m = -1e30f;
    for (int d = 0; d < 32; ++d) m = fmaxf(m, S[tid * 33 + d]);
    float sum = 0.f;
    for (int d = 0; d < 32; ++d) sum += __expf(S[tid * 33 + d] - m);
    float inv = 1.0f / sum;
    for (int d = 0; d < 32; ++d)
      attn[(size_t)bh * 1024 + tid * 32 + d] = __expf(S[tid * 33 + d] - m) * inv;
  }
}

// out2[bh][c][n] = sum_d attn[bh][c][d] * v[bh][d][n]   (M=32, K=32, N=16384)
__global__ __launch_bounds__(256)
void av_gemm(const float* __restrict__ attn, const float* __restrict__ v,
             float* __restrict__ out2) {
  __shared__ float As[32 * 33];
  const int tid  = threadIdx.x;
  const int lane = tid & 31;
  const int w    = tid >> 5;
  const int bh = blockIdx.y;
  const int b  = bh / NH_, h = bh - b * NH_;
  const int nBlock = blockIdx.x * 256;

  for (int i = tid; i < 1024; i += 256)
    As[(i >> 5) * 33 + (i & 31)] = attn[(size_t)bh * 1024 + i];
  __syncthreads();

  const int wm = w >> 2, wn = w & 3;
  const int mBase = wm * 16;
  const int nBase = nBlock + wn * 64;
  const int hl = lane >> 4, l15 = lane & 15;
  const float* vB = v + ((size_t)b * C_ + h * CH_) * HW_;

  v8f acc0 = {}, acc1 = {}, acc2 = {}, acc3 = {};
#pragma unroll
  for (int k0 = 0; k0 < 32; k0 += 4) {
    const int ks = k0 + 2 * hl;
    v2f a;
    a.x = As[(mBase + l15) * 33 + ks];
    a.y = As[(mBase + l15) * 33 + ks + 1];
    const float* p = vB + (size_t)ks * HW_ + nBase + l15;
    v2f b0, b1, b2, b3;
    b0.x = p[0];  b0.y = p[HW_];
    b1.x = p[16]; b1.y = p[HW_ + 16];
    b2.x = p[32]; b2.y = p[HW_ + 32];
    b3.x = p[48]; b3.y = p[HW_ + 48];
    acc0 = wmma_f32(a, b0, acc0);
    acc1 = wmma_f32(a, b1, acc1);
    acc2 = wmma_f32(a, b2, acc2);
    acc3 = wmma_f32(a, b3, acc3);
  }
  float* oB = out2 + ((size_t)b * C_ + h * CH_) * HW_;
  const int rowBase = mBase + hl * 8;
#pragma unroll
  for (int vv = 0; vv < 8; ++vv) {
    float* dp = oB + (size_t)(rowBase + vv) * HW_ + nBase + l15;
    dp[0]  = acc0[vv];
    dp[16] = acc1[vv];
    dp[32] = acc2[vv];
    dp[48] = acc3[vv];
  }
}

extern "C" void kernel_launch(void* const* d_in, const int* in_sizes, int n_in,
                              void* d_out, int out_size, void* d_ws, size_t ws_size,
                              hipStream_t stream) {
  const float* x    = (const float*)d_in[0];
  const float* y    = (const float*)d_in[1];
  const float* Wq   = (const float*)d_in[2];
  const float* Wk   = (const float*)d_in[3];
  const float* Wv   = (const float*)d_in[4];
  const float* Wo   = (const float*)d_in[5];
  const float* temp = (const float*)d_in[6];
  // d_in[7] = num_heads (compile-time constant NH_ here)
  (void)in_sizes; (void)n_in; (void)out_size; (void)ws_size;

  const size_t BCHW = (size_t)B_ * C_ * HW_;
  float* q        = (float*)d_ws;
  float* k        = q + BCHW;
  float* v        = k + BCHW;
  float* inv_norm = v + BCHW;                 // 2*B*C floats
  float* attn     = inv_norm + 2 * B_ * C_;   // B*NH*32*32 floats
  float* out2     = q;                        // q is dead after attn_kernel

  dim3 gGemm(HW_ / 256, C_ / 32, B_);
  proj_gemm<<<gGemm, 256, 0, stream>>>(Wq, x, q);
  proj_gemm<<<gGemm, 256, 0, stream>>>(Wk, y, k);
  proj_gemm<<<gGemm, 256, 0, stream>>>(Wv, y, v);
  row_norms<<<2 * B_ * C_, 256, 0, stream>>>(q, k, inv_norm);
  attn_kernel<<<B_ * NH_, 256, 0, stream>>>(q, k, inv_norm, temp, attn);
  av_gemm<<<dim3(HW_ / 256, B_ * NH_), 256, 0, stream>>>(attn, v, out2);
  proj_gemm<<<gGemm, 256, 0, stream>>>(Wo, out2, (float*)d_out);
}